// MoEsparseRouting_13030930776374
// MI455X (gfx1250) — compile-verified
//
#include <hip/hip_runtime.h>

// TT-MoE forward for MI455X (gfx1250), wave32, WMMA f32 16x16x4.
//
// Reference pipeline fused into 4 per-batch GEMMs:
//   C[64,8]   = fuse(core_first, mid0)        G1: [S*64,64] x C  (K=64)
//   D[512,8]  = fuse(mid1, mid2)              G2: [S,512]  x D  (K=512)
//   E[8,512]  = fuse(mid3, mid4)              G3: [S,8]    x E  (K=8)
//   F[8,64]   = fuse(mid5, core_last)         G4: [S*64,8] x F  (K=8)
// Memory-bound (128 MB HBM traffic, ~0.6 GFLOP) -> fp32 WMMA, all
// intermediates in LDS, single fused compute kernel.
// Lane masking done via mask-multiply on clamped addresses (no EXEC-masked
// LDS loads); loop-invariant WMMA fragments hoisted out of tile loops.

typedef __attribute__((ext_vector_type(2))) float v2f;
typedef __attribute__((ext_vector_type(8))) float v8f;

#define WS_C 0        // B*512
#define WS_D 4096     // B*4096
#define WS_E 36864    // B*4096
#define WS_F 69632    // B*512
#define TOK  8        // tokens per block in main kernel

__device__ __forceinline__ v8f wmma4(v2f a, v2f b, v8f c) {
  // D = A(16x4 f32) * B(4x16 f32) + C(16x16 f32)
  return __builtin_amdgcn_wmma_f32_16x16x4_f32(
      /*neg_a=*/false, a, /*neg_b=*/false, b,
      /*c_mod=*/(short)0, c, /*reuse_a=*/false, /*reuse_b=*/false);
}

// ---------------------------------------------------------------------------
// Prep: gate-merge cores per batch, build fused matrices C, D, E, F.
// ---------------------------------------------------------------------------
__global__ __launch_bounds__(256)
void tt_prep_kernel(const float* __restrict__ gates,
                    const float* __restrict__ core_first,
                    const float* __restrict__ cores_mid,
                    const float* __restrict__ core_last,
                    float* __restrict__ ws) {
  const int b = blockIdx.x;
  const int tid = threadIdx.x;
  __shared__ float g[8];
  __shared__ float Wf[64];      // [m0*8 + r]
  __shared__ float Wm[6][512];  // [k][r*64 + m*8 + p]
  __shared__ float Wl[64];      // [r*8 + n3]

  if (tid < 8) g[tid] = gates[b * 8 + tid];
  __syncthreads();

  if (tid < 64) {
    float s0 = 0.f, s1 = 0.f;
#pragma unroll
    for (int e = 0; e < 8; ++e) {
      s0 += g[e] * core_first[e * 64 + tid];
      s1 += g[e] * core_last[e * 64 + tid];
    }
    Wf[tid] = s0;
    Wl[tid] = s1;
  }
  for (int idx = tid; idx < 3072; idx += 256) {
    const int k = idx >> 9, j = idx & 511;
    float s = 0.f;
#pragma unroll
    for (int e = 0; e < 8; ++e) s += g[e] * cores_mid[(k * 8 + e) * 512 + j];
    Wm[k][j] = s;
  }
  __syncthreads();

  // C[(f3*8+f4)*8 + p2] = sum_r Wf[f4,r] * Wm0[r,f3,p2]
  for (int idx = tid; idx < 512; idx += 256) {
    const int f3 = idx >> 6, f4 = (idx >> 3) & 7, p2 = idx & 7;
    float s = 0.f;
#pragma unroll
    for (int r = 0; r < 8; ++r) s += Wf[f4 * 8 + r] * Wm[0][r * 64 + f3 * 8 + p2];
    ws[WS_C + b * 512 + idx] = s;
  }
  // D[(p2*64+f1*8+f2)*8 + p4] = sum_p3 Wm1[p2,f2,p3] * Wm2[p3,f1,p4]
  for (int idx = tid; idx < 4096; idx += 256) {
    const int p2 = idx >> 9, f1 = (idx >> 6) & 7, f2 = (idx >> 3) & 7, p4 = idx & 7;
    float s = 0.f;
#pragma unroll
    for (int p3 = 0; p3 < 8; ++p3)
      s += Wm[1][p2 * 64 + f2 * 8 + p3] * Wm[2][p3 * 64 + f1 * 8 + p4];
    ws[WS_D + b * 4096 + idx] = s;
  }
  // E[p4*512 + (n0*8+n1)*8 + p6] = sum_p5 Wm3[p4,n0,p5] * Wm4[p5,n1,p6]
  for (int idx = tid; idx < 4096; idx += 256) {
    const int p4 = idx >> 9, n0 = (idx >> 6) & 7, n1 = (idx >> 3) & 7, p6 = idx & 7;
    float s = 0.f;
#pragma unroll
    for (int p5 = 0; p5 < 8; ++p5)
      s += Wm[3][p4 * 64 + n0 * 8 + p5] * Wm[4][p5 * 64 + n1 * 8 + p6];
    ws[WS_E + b * 4096 + idx] = s;
  }
  // F[p6*64 + n2*8 + n3] = sum_p7 Wm5[p6,n2,p7] * Wl[p7,n3]
  for (int idx = tid; idx < 512; idx += 256) {
    const int p6 = idx >> 6, n2 = (idx >> 3) & 7, n3 = idx & 7;
    float s = 0.f;
#pragma unroll
    for (int p7 = 0; p7 < 8; ++p7)
      s += Wm[5][p6 * 64 + n2 * 8 + p7] * Wl[p7 * 8 + n3];
    ws[WS_F + b * 512 + idx] = s;
  }
}

// ---------------------------------------------------------------------------
// Fully fused token pipeline: G1 -> G2 -> G3 -> G4 (all WMMA f32 16x16x4).
// Block = (batch b, 8 tokens), 128 threads = 4 wave32.
// ---------------------------------------------------------------------------
__global__ __launch_bounds__(128)
void tt_main_kernel(const float* __restrict__ X,
                    const float* __restrict__ ws,
                    float* __restrict__ out) {
  const int b    = blockIdx.y;
  const int s0   = blockIdx.x * TOK;
  const int tid  = threadIdx.x;
  const int wave = tid >> 5;
  const int lane = tid & 31;
  const int half = lane >> 4;   // lane 0-15: K{0,1} / row v ; lane 16-31: K{2,3} / row v+8
  const int l15  = lane & 15;
  const int   c7    = l15 & 7;                    // clamped column (always in-bounds)
  const float maskN = (l15 < 8) ? 1.f : 0.f;      // valid N column (B pads N=8 -> 16)
  const float maskA = (l15 < TOK) ? 1.f : 0.f;    // valid token row in small GEMMs

  __shared__ float sC[512];
  __shared__ float sD[4096];
  __shared__ float sE[4096];
  __shared__ float sF[512];
  __shared__ float sTT[TOK * 512];  // t2, then reused as t6
  __shared__ float sPart[4][128];   // per-wave G2 partials [tok*8+p4]
  __shared__ float sT4[64];         // t4 [tok*8+p4]

  for (int i = tid; i < 512; i += 128)  sC[i] = ws[WS_C + b * 512 + i];
  for (int i = tid; i < 4096; i += 128) sD[i] = ws[WS_D + b * 4096 + i];
  for (int i = tid; i < 4096; i += 128) sE[i] = ws[WS_E + b * 4096 + i];
  for (int i = tid; i < 512; i += 128)  sF[i] = ws[WS_F + b * 512 + i];
  __syncthreads();

  const float* Xb = X + (size_t)(b * 512 + s0) * 4096;

  // ---- G1: rows (tok,f1,f2) = 512, K=64, N=8 (padded 16) -> t2 ----
  // B fragments (matrix C) are identical for every row-tile: hoist all 16.
  v2f bC[16];
#pragma unroll
  for (int kk = 0; kk < 16; ++kk) {
    const int kb = kk * 4 + half * 2;
    bC[kk].x = maskN * sC[(kb + 0) * 8 + c7];
    bC[kk].y = maskN * sC[(kb + 1) * 8 + c7];
  }
  for (int t = wave; t < 32; t += 4) {
    const int arow = t * 16 + l15;
    const int tok = arow >> 6, f1f2 = arow & 63;
    const float* arowp = Xb + (size_t)tok * 4096 + f1f2 * 64 + half * 2;
    v8f acc = {};
#pragma unroll
    for (int kk = 0; kk < 16; ++kk) {
      v2f a = *(const v2f*)(arowp + kk * 4);
      acc = wmma4(a, bC[kk], acc);
    }
    if (l15 < 8) {  // p2 = l15; store t2[token][p2*64 + f1f2]
#pragma unroll
      for (int v = 0; v < 8; ++v) {
        const int row = t * 16 + v + half * 8;
        sTT[(row >> 6) * 512 + l15 * 64 + (row & 63)] = acc[v];
      }
    }
  }
  __syncthreads();

  // ---- G2: t2[8tok,512] x D[512,8], K split across 4 waves ----
  {
    v8f acc = {};
    for (int kk = 0; kk < 32; ++kk) {
      const int kb = wave * 128 + kk * 4 + half * 2;
      v2f a = *(const v2f*)(&sTT[c7 * 512 + kb]);
      a = a * maskA;
      v2f bf;
      bf.x = maskN * sD[(kb + 0) * 8 + c7];
      bf.y = maskN * sD[(kb + 1) * 8 + c7];
      acc = wmma4(a, bf, acc);
    }
    if (half == 0 && l15 < 8) {  // rows v=0..7 (tokens), cols l15=0..7 (p4)
#pragma unroll
      for (int v = 0; v < 8; ++v) sPart[wave][v * 8 + l15] = acc[v];
    }
    __syncthreads();
    if (tid < 64)
      sT4[tid] = sPart[0][tid] + sPart[1][tid] + sPart[2][tid] + sPart[3][tid];
    __syncthreads();
  }

  // ---- G3: t4[8tok,8] x E[8,512] -> t6 (reuse sTT) ----
  // A fragments are tile-invariant: hoist both k-steps.
  v2f a3[2];
#pragma unroll
  for (int kk = 0; kk < 2; ++kk) {
    const int kb = kk * 4 + half * 2;
    v2f a = *(const v2f*)(&sT4[c7 * 8 + kb]);
    a3[kk] = a * maskA;
  }
  for (int t = wave; t < 32; t += 4) {
    v8f acc = {};
#pragma unroll
    for (int kk = 0; kk < 2; ++kk) {
      const int kb = kk * 4 + half * 2;
      v2f bf;
      bf.x = sE[(kb + 0) * 512 + t * 16 + l15];
      bf.y = sE[(kb + 1) * 512 + t * 16 + l15];
      acc = wmma4(a3[kk], bf, acc);
    }
    if (half == 0) {  // rows v = token 0..7; t6[token][(n0n1)*8+p6] = col
#pragma unroll
      for (int v = 0; v < 8; ++v) sTT[v * 512 + t * 16 + l15] = acc[v];
    }
  }
  __syncthreads();

  // ---- G4: rows (tok,n0,n1)=512, K=8, N=64 -> out (contiguous 64/row) ----
  // Loop column tile outermost (uniform) so F fragments hoist per ct.
  float* outB = out + (size_t)(b * 512 + s0) * 4096;
#pragma unroll
  for (int ct = 0; ct < 4; ++ct) {
    v2f bF[2];
#pragma unroll
    for (int kk = 0; kk < 2; ++kk) {
      const int kb = kk * 4 + half * 2;
      bF[kk].x = sF[(kb + 0) * 64 + ct * 16 + l15];
      bF[kk].y = sF[(kb + 1) * 64 + ct * 16 + l15];
    }
    for (int rt = wave; rt < 32; rt += 4) {
      const int arow = rt * 16 + l15;
      const int tok = arow >> 6, nn = arow & 63;
      v8f acc = {};
#pragma unroll
      for (int kk = 0; kk < 2; ++kk) {
        const int kb = kk * 4 + half * 2;
        v2f a = *(const v2f*)(&sTT[tok * 512 + nn * 8 + kb]);
        acc = wmma4(a, bF[kk], acc);
      }
#pragma unroll
      for (int v = 0; v < 8; ++v) {
        const int row = rt * 16 + v + half * 8;
        outB[(size_t)(row >> 6) * 4096 + (row & 63) * 64 + ct * 16 + l15] = acc[v];
      }
    }
  }
}

extern "C" void kernel_launch(void* const* d_in, const int* in_sizes, int n_in,
                              void* d_out, int out_size, void* d_ws, size_t ws_size,
                              hipStream_t stream) {
  const float* X          = (const float*)d_in[0];
  const float* gates      = (const float*)d_in[1];
  const float* core_first = (const float*)d_in[2];
  const float* cores_mid  = (const float*)d_in[3];
  const float* core_last  = (const float*)d_in[4];
  float* ws  = (float*)d_ws;
  float* out = (float*)d_out;

  tt_prep_kernel<<<8, 256, 0, stream>>>(gates, core_first, cores_mid, core_last, ws);
  tt_main_kernel<<<dim3(64, 8), 128, 0, stream>>>(X, ws, out);
}